// BallQuery_16346645529138
// MI455X (gfx1250) — compile-verified
//
#include <hip/hip_runtime.h>

typedef __attribute__((ext_vector_type(2))) float v2f;
typedef __attribute__((ext_vector_type(8))) float v8f;

#define BQ_B        8
#define BQ_N        8192
#define BQ_M        2048
#define BQ_NSAMPLE  32
#define BQ_RADIUS2  0.01f
#define CHUNK       1024          // points staged in LDS per outer iter (16KB)
#define THREADS     128           // 4 wave32s
#define QPW         16            // queries per wave
#define QPB         64            // queries per block

__global__ __launch_bounds__(THREADS)
void ball_query_wmma(const float* __restrict__ xyz,
                     const float* __restrict__ new_xyz,
                     int* __restrict__ out)
{
    __shared__ float4 pts[CHUNK];          // (px,py,pz,|p|^2)
    __shared__ float  thr_lds[THREADS/32][QPW];

    const int tid   = threadIdx.x;
    const int lane  = tid & 31;
    const int wave  = tid >> 5;
    const int row16 = lane & 15;
    const int half  = lane >> 4;           // 0: lanes 0-15 (K=0,1), 1: lanes 16-31 (K=2,3)

    const int tilesPerBatch = BQ_M / QPB;  // 32
    const int b     = blockIdx.x / tilesPerBatch;
    const int qtile = blockIdx.x % tilesPerBatch;
    const int qrow  = qtile * QPB + wave * QPW + row16;

    // ---- query -> A fragment (-2qx,-2qy | -2qz, 1) and per-row threshold ----
    const float* q = new_xyz + ((size_t)b * BQ_M + qrow) * 3;
    const float qx = q[0], qy = q[1], qz = q[2];
    v2f avec;
    avec.x = half ? (-2.0f * qz) : (-2.0f * qx);
    avec.y = half ? 1.0f         : (-2.0f * qy);
    if (half == 0) thr_lds[wave][row16] = BQ_RADIUS2 - (qx*qx + qy*qy + qz*qz);
    __syncthreads();

    // acc[v] (this lane) belongs to row M = v + 8*half -> preload its threshold
    float thr[8];
    #pragma unroll
    for (int v = 0; v < 8; ++v) thr[v] = thr_lds[wave][v + 8 * half];

    int  cnt   = (half == 0) ? 0 : BQ_NSAMPLE;   // upper lanes own no row
    int  first = 0;
    int* outp  = out + ((size_t)b * BQ_M + qrow) * BQ_NSAMPLE;

    const float* xb = xyz + (size_t)b * BQ_N * 3;

    for (int base = 0; base < BQ_N; base += CHUNK) {
        // cooperative stage: CHUNK points -> LDS with precomputed |p|^2
        for (int i = tid; i < CHUNK; i += THREADS) {
            const float* p = xb + (size_t)(base + i) * 3;
            const float px = p[0], py = p[1], pz = p[2];
            pts[i] = make_float4(px, py, pz, px*px + py*py + pz*pz);
        }
        __syncthreads();

        // uniform per-wave early-exit check
        if (__builtin_amdgcn_ballot_w32(cnt >= BQ_NSAMPLE) != 0xFFFFFFFFu) {
            for (int c = 0; c < CHUNK; c += 16) {
                // B fragment: lane = column N, halves supply (px,py)/(pz,|p|^2)
                const float4 pv = pts[c + row16];
                v2f bvec;
                bvec.x = half ? pv.z : pv.x;
                bvec.y = half ? pv.w : pv.y;

                v8f acc = {0.f,0.f,0.f,0.f,0.f,0.f,0.f,0.f};
                // acc = A*B : |p|^2 - 2 q.p  for a 16x16 (query x point) tile
                acc = __builtin_amdgcn_wmma_f32_16x16x4_f32(
                        false, avec, false, bvec, (short)0, acc, false, false);

                // 8 ballots -> per-row 16-bit hit masks; lane r (<16) owns row r
                unsigned mymask = 0u;
                #pragma unroll
                for (int v = 0; v < 8; ++v) {
                    const unsigned bal =
                        __builtin_amdgcn_ballot_w32(acc[v] < thr[v]);
                    if (half == 0 && (row16 & 7) == v)
                        mymask = (row16 < 8) ? (bal & 0xFFFFu) : (bal >> 16);
                }

                // serial residue: emit indices of set bits in ascending order
                while (mymask != 0u && cnt < BQ_NSAMPLE) {
                    const int bit = __builtin_ctz(mymask);
                    const int idx = base + c + bit;
                    if (cnt == 0) first = idx;
                    outp[cnt] = idx;
                    ++cnt;
                    mymask &= (mymask - 1u);
                }
            }
        }
        __syncthreads();
    }

    // pad: no hits -> sentinel n+1 everywhere; else replicate first hit
    if (half == 0) {
        const int fill = (cnt == 0) ? (BQ_N + 1) : first;
        for (int j = cnt; j < BQ_NSAMPLE; ++j) outp[j] = fill;
    }
}

extern "C" void kernel_launch(void* const* d_in, const int* in_sizes, int n_in,
                              void* d_out, int out_size, void* d_ws, size_t ws_size,
                              hipStream_t stream) {
    (void)in_sizes; (void)n_in; (void)d_ws; (void)ws_size; (void)out_size;
    const float* xyz     = (const float*)d_in[0];   // (8, 8192, 3) f32
    const float* new_xyz = (const float*)d_in[1];   // (8, 2048, 3) f32
    int* out = (int*)d_out;                         // (8, 2048, 32) i32

    const int blocks = BQ_B * (BQ_M / QPB);         // 8 * 32 = 256
    ball_query_wmma<<<blocks, THREADS, 0, stream>>>(xyz, new_xyz, out);
}